// RandomShiftModule_32195074850843
// MI455X (gfx1250) — compile-verified
//
#include <hip/hip_runtime.h>

// RandomShift: out[n,c,y,x] = in[n,c, clamp(y+hs-PAD), clamp(x+ws-PAD)]
// Pure streaming gather: ~403 MB HBM traffic, 0 FLOPs -> bandwidth-bound
// (floor ~17us @ 23.3 TB/s on MI455X).
//
// One wave32 per (n, y) pair, processing ALL 12 channels: the shifts /
// clamped indices are identical across channels, and the per-channel plane
// offset (c * 64 KiB) fits the 24-bit vmem immediate offset, so the unrolled
// loop is a dense clause of NT dword loads + NT b128 stores off one base
// address. 12 KB of HBM traffic per wave; index math amortized 12x.

#define PAD_ 4
#define C_   12
#define H_   128
#define W_   128

typedef __attribute__((ext_vector_type(4))) float v4f;

__global__ __launch_bounds__(256) void random_shift_kernel(
    const float* __restrict__ x,
    const int*   __restrict__ h_starts,
    const int*   __restrict__ w_starts,
    float*       __restrict__ out)
{
    const int lane = threadIdx.x & 31;          // lane within wave32
    const int wid  = threadIdx.x >> 5;          // wave within block (0..7)
    const int r    = blockIdx.x * 8 + wid;      // (n, y) pair id, 0..N*H-1

    const int y = r & (H_ - 1);
    const int n = r >> 7;                       // r / H_

    // per-sample shifts (wave-uniform, loaded once per 12 rows)
    const int hs = h_starts[n];
    const int ws = w_starts[n];
    const int ysrc = min(max(y + hs - PAD_, 0), H_ - 1);
    const int dw   = ws - PAD_;                 // in [-4, 4]

    const int x0 = lane << 2;                   // 4 floats/lane; dst 16B aligned

    // branchless edge-replicate clamp; shared by all 12 channels
    const int i0 = min(max(x0 + 0 + dw, 0), W_ - 1);
    const int i1 = min(max(x0 + 1 + dw, 0), W_ - 1);
    const int i2 = min(max(x0 + 2 + dw, 0), W_ - 1);
    const int i3 = min(max(x0 + 3 + dw, 0), W_ - 1);

    const size_t nbase = (size_t)n * (C_ * H_ * W_);
    const float* __restrict__ src = x   + nbase + (size_t)ysrc * W_;
    float*       __restrict__ dst = out + nbase + (size_t)y    * W_;

    // All 12 channel planes: offset c*H*W*4 = c*65536 B fits the 24-bit
    // vmem IOFFSET -> one base VGPR address, immediate-offset clauses.
#pragma unroll
    for (int c = 0; c < C_; ++c) {
        const int off = c * (H_ * W_);
        v4f v;
        v.x = __builtin_nontemporal_load(src + off + i0); // streaming reads
        v.y = __builtin_nontemporal_load(src + off + i1);
        v.z = __builtin_nontemporal_load(src + off + i2);
        v.w = __builtin_nontemporal_load(src + off + i3);
        __builtin_nontemporal_store(v, (v4f*)(dst + off + x0)); // b128 NT
    }
}

extern "C" void kernel_launch(void* const* d_in, const int* in_sizes, int n_in,
                              void* d_out, int out_size, void* d_ws, size_t ws_size,
                              hipStream_t stream) {
    (void)in_sizes; (void)n_in; (void)d_ws; (void)ws_size; (void)out_size;

    const float* x        = (const float*)d_in[0];
    const int*   h_starts = (const int*)d_in[1];
    const int*   w_starts = (const int*)d_in[2];
    float*       out      = (float*)d_out;

    const int pairs  = 256 * H_;        // N*H = 32768 (n,y) pairs
    const int blocks = pairs / 8;       // 8 waves per 256-thread block

    random_shift_kernel<<<blocks, 256, 0, stream>>>(x, h_starts, w_starts, out);
}